// HausdorffDTLoss_13331578486851
// MI455X (gfx1250) — compile-verified
//
#include <hip/hip_runtime.h>

#define H 256
#define W 256
#define PLANE (H * W)        // 65536
#define NIMG 16              // 8 pred + 8 target images
#define NTOT (8 * PLANE)     // 524288 elements in final mean
#define BIGF 1e10f
#define BLOCKS_C 128

typedef __attribute__((ext_vector_type(2))) float v2f;
typedef __attribute__((ext_vector_type(8))) float v8f;

#if __has_builtin(__builtin_amdgcn_wmma_f32_16x16x4_f32)
#define USE_WMMA 1
#else
#define USE_WMMA 0
#endif

#if __has_builtin(__builtin_amdgcn_global_load_async_to_lds_b32)
#define USE_ASYNC 1
#else
#define USE_ASYNC 0
#endif

typedef __attribute__((address_space(3))) int lds_int;

// ---------------------------------------------------------------------------
// Zero the per-image fg.any() flags (atomicOr target) each call.
// ---------------------------------------------------------------------------
__global__ void hdt_init(int* __restrict__ flags) {
    int t = threadIdx.x;
    if (t < NIMG) flags[t] = 0;
}

// ---------------------------------------------------------------------------
// Column pass: per-column 1D distance (in rows) to nearest background pixel,
// for BOTH masks (fg plane: bg = !fg; bg plane: bg = fg).  Writes g^2.
// Down-scan writes provisional up-distance (0 encodes "not in mask"),
// up-scan finishes min(up, down, BIG) and squares.
// One thread per column; 16 blocks x 256 threads.
// ---------------------------------------------------------------------------
__global__ void hdt_colpass(const float* __restrict__ pred,
                            const float* __restrict__ target,
                            float* __restrict__ g2,
                            int* __restrict__ flags) {
    const int m = blockIdx.x;   // image 0..15 (0..7 pred, 8..15 target)
    const int c = threadIdx.x;  // column 0..255
    const float* img = (m < 8) ? (pred + (size_t)m * PLANE)
                               : (target + (size_t)(m - 8) * PLANE);
    float* pf = g2 + (size_t)(2 * m) * PLANE;      // fg-mask plane
    float* pb = g2 + (size_t)(2 * m + 1) * PLANE;  // bg-mask plane

    int lastBg = -1, lastFg = -1;
    int anyFg = 0;
    for (int i = 0; i < H; ++i) {
        float v = img[i * W + c];
        bool fg = v > 0.5f;
        anyFg |= fg ? 1 : 0;
        if (fg) lastFg = i; else lastBg = i;
        // provisional up-distance; 0 <=> pixel not in that plane's mask
        float dF = fg ? ((lastBg >= 0) ? (float)(i - lastBg) : BIGF) : 0.0f;
        float dB = fg ? 0.0f : ((lastFg >= 0) ? (float)(i - lastFg) : BIGF);
        pf[i * W + c] = dF;
        pb[i * W + c] = dB;
    }
    if (anyFg) atomicOr(&flags[m], 1);

    int nextBg = -1, nextFg = -1;
    for (int i = H - 1; i >= 0; --i) {
        float tF = pf[i * W + c];
        float tB = pb[i * W + c];
        bool fg = tF > 0.0f;   // encoding from down-scan
        float gF, gB;
        if (fg) {
            float d = tF;
            if (nextBg >= 0) d = fminf(d, (float)(nextBg - i));
            gF = fminf(d, BIGF);
            gB = 0.0f;
            nextFg = i;
        } else {
            float d = tB;
            if (nextFg >= 0) d = fminf(d, (float)(nextFg - i));
            gB = fminf(d, BIGF);
            gF = 0.0f;
            nextBg = i;
        }
        pf[i * W + c] = gF * gF;
        pb[i * W + c] = gB * gB;
    }
}

// ---------------------------------------------------------------------------
// Row pass: exact separable EDT:  d2[j] = min_k g2[k] + (j-k)^2  for both
// planes of one row; writes dt2 = d2_fg + d2_bg (zeroed if image had no fg)
// in place over the fg plane.  One block per (row, image).
// LDS staging uses CDNA5 async global->LDS loads when available.
// ---------------------------------------------------------------------------
__global__ void hdt_rowpass(float* __restrict__ g2,
                            const int* __restrict__ flags) {
    __shared__ float sFg[W];
    __shared__ float sBg[W];
    const int row = blockIdx.x;
    const int m   = blockIdx.y;
    const int j   = threadIdx.x;

    float* pf = g2 + (size_t)(2 * m) * PLANE + (size_t)row * W;
    float* pb = g2 + (size_t)(2 * m + 1) * PLANE + (size_t)row * W;

#if USE_ASYNC
    __builtin_amdgcn_global_load_async_to_lds_b32(
        (int*)(pf + j), (lds_int*)&sFg[j], 0, 0);
    __builtin_amdgcn_global_load_async_to_lds_b32(
        (int*)(pb + j), (lds_int*)&sBg[j], 0, 0);
#if __has_builtin(__builtin_amdgcn_s_wait_asynccnt)
    __builtin_amdgcn_s_wait_asynccnt(0);
#else
    asm volatile("s_wait_asynccnt 0" ::: "memory");
#endif
    __syncthreads();
#else
    sFg[j] = pf[j];
    sBg[j] = pb[j];
    __syncthreads();
#endif

    float bf = 3.0e38f, bb = 3.0e38f;
#pragma unroll 4
    for (int k = 0; k < W; ++k) {
        float diff = (float)(j - k);
        float off = diff * diff;             // exact: |j-k| <= 255
        bf = fminf(bf, sFg[k] + off);
        bb = fminf(bb, sBg[k] + off);
    }
    float any = (flags[m] != 0) ? 1.0f : 0.0f;
    pf[j] = (bf + bb) * any;                 // dt2 for this image
}

// ---------------------------------------------------------------------------
// Reduction stage 1: w = (pred - target)^2 * (dt2_pred + dt2_target),
// accumulated via v_wmma_f32_16x16x4_f32 with a ones B-matrix (each WMMA
// folds 64 new values into the f32 accumulator; the 16 output columns are
// duplicates -> /16 at the end).  128 blocks x 256 threads, 4096 elems/block.
// ---------------------------------------------------------------------------
__device__ __forceinline__ float hdt_wval(int e,
                                          const float* __restrict__ pred,
                                          const float* __restrict__ target,
                                          const float* __restrict__ g2) {
    int b = e >> 16;          // batch
    int p = e & 0xFFFF;       // pixel
    float d = pred[e] - target[e];
    float dist = g2[(size_t)(2 * b) * PLANE + p] +
                 g2[(size_t)(2 * (b + 8)) * PLANE + p];
    return d * d * dist;
}

__global__ void hdt_reduce1(const float* __restrict__ pred,
                            const float* __restrict__ target,
                            const float* __restrict__ g2,
                            float* __restrict__ partials) {
    __shared__ float red[256];
    const int tid  = threadIdx.x;
    const int lane = tid & 31;
    const int wave = tid >> 5;
    const int base = blockIdx.x * 4096 + wave * 512;

#if USE_WMMA
    v8f c = {};
    v2f bones; bones[0] = 1.0f; bones[1] = 1.0f;
#else
    float acc = 0.0f;
#endif
#pragma unroll
    for (int t = 0; t < 8; ++t) {
        int e = base + t * 64 + lane * 2;
        float w0 = hdt_wval(e,     pred, target, g2);
        float w1 = hdt_wval(e + 1, pred, target, g2);
#if USE_WMMA
        v2f a; a[0] = w0; a[1] = w1;
        c = __builtin_amdgcn_wmma_f32_16x16x4_f32(
                /*neg_a=*/false, a, /*neg_b=*/false, bones,
                /*c_mod=*/(short)0, c, /*reuse_a=*/false, /*reuse_b=*/false);
#else
        acc += w0 + w1;
#endif
    }
    float local;
#if USE_WMMA
    local = 0.0f;
#pragma unroll
    for (int q = 0; q < 8; ++q) local += c[q];
#else
    local = acc;
#endif
    red[tid] = local;
    __syncthreads();
#pragma unroll
    for (int s = 128; s > 0; s >>= 1) {
        if (tid < s) red[tid] += red[tid + s];
        __syncthreads();
    }
    if (tid == 0) partials[blockIdx.x] = red[0];
}

// ---------------------------------------------------------------------------
// Reduction stage 2: single block folds the 128 partials -> scalar mean.
// ---------------------------------------------------------------------------
__global__ void hdt_reduce2(const float* __restrict__ partials,
                            float* __restrict__ out) {
    __shared__ float red[BLOCKS_C];
    const int tid = threadIdx.x;  // 128 threads
    red[tid] = partials[tid];
    __syncthreads();
#pragma unroll
    for (int s = BLOCKS_C / 2; s > 0; s >>= 1) {
        if (tid < s) red[tid] += red[tid + s];
        __syncthreads();
    }
    if (tid == 0) {
        const float dup = USE_WMMA ? (1.0f / 16.0f) : 1.0f;
        out[0] = red[0] * dup * (1.0f / (float)NTOT);
    }
}

// ---------------------------------------------------------------------------
extern "C" void kernel_launch(void* const* d_in, const int* in_sizes, int n_in,
                              void* d_out, int out_size, void* d_ws, size_t ws_size,
                              hipStream_t stream) {
    const float* pred   = (const float*)d_in[0];
    const float* target = (const float*)d_in[1];
    float* out = (float*)d_out;

    char* ws = (char*)d_ws;
    float* g2 = (float*)ws;                                  // 32 planes * 256KB = 8 MB
    size_t g2_bytes = (size_t)2 * NIMG * PLANE * sizeof(float);
    int*   flags    = (int*)(ws + g2_bytes);                 // 16 ints
    float* partials = (float*)(ws + g2_bytes + 256);         // 128 floats

    hipLaunchKernelGGL(hdt_init,    dim3(1),        dim3(32),  0, stream, flags);
    hipLaunchKernelGGL(hdt_colpass, dim3(NIMG),     dim3(W),   0, stream,
                       pred, target, g2, flags);
    hipLaunchKernelGGL(hdt_rowpass, dim3(H, NIMG),  dim3(W),   0, stream, g2, flags);
    hipLaunchKernelGGL(hdt_reduce1, dim3(BLOCKS_C), dim3(256), 0, stream,
                       pred, target, g2, partials);
    hipLaunchKernelGGL(hdt_reduce2, dim3(1), dim3(BLOCKS_C),   0, stream,
                       partials, out);
}